// MultiScaleGraphConstruction_41120016892412
// MI455X (gfx1250) — compile-verified
//
#include <hip/hip_runtime.h>

typedef __attribute__((ext_vector_type(2))) float v2f;
typedef __attribute__((ext_vector_type(8))) float v8f;

#define NPTS   2048
#define NBATCH 8
#define NTILE  128                              // NPTS / 16
#define PLANE  ((size_t)NBATCH * NPTS * NPTS)   // elements per scale

// One wave32 computes one 16x16 tile of dist^2 directly in the WMMA
// accumulator using the spare K slot (K=4, coords use 3):
//   WMMA1: A=(-2x,-2y,-2z,|row|^2), B=(x,y,z,1)   -> -2*dot + |row_m|^2
//   WMMA2: A=(1,0,0,0),            B0=|col|^2     -> += |col_n|^2
// Only two cross-half shfl_xor(16) needed to assemble the norms.
__global__ __launch_bounds__(256)
void msgc_dist_kernel(const float* __restrict__ pos, float* __restrict__ out)
{
    const int lane = threadIdx.x & 31;
    const int wave = threadIdx.x >> 5;
    const int tile = blockIdx.x * 8 + wave;        // 131072 tiles total

    const int b  = tile >> 14;                     // / (128*128)
    const int tr = (tile >> 7) & 127;
    const int tc = tile & 127;

    const int half = lane >> 4;                    // 0: K=0,1   1: K=2,3
    const int l15  = lane & 15;

    const float* pb = pos + (size_t)b * NPTS * 3;
    const float* pr = pb + (size_t)(tr * 16 + l15) * 3;
    const float* pc = pb + (size_t)(tc * 16 + l15) * 3;

    // Row point: lanes 0-15 load (x,y), lanes 16-31 load z.
    float r0, r1;
    if (half == 0) { r0 = pr[0]; r1 = pr[1]; }
    else           { r0 = pr[2]; r1 = 0.0f; }
    // Col point likewise.
    float c0, c1;
    if (half == 0) { c0 = pc[0]; c1 = pc[1]; }
    else           { c0 = pc[2]; c1 = 0.0f; }

    // Fold the two lane-halves to get full squared norms in every lane.
    float rp = r0 * r0 + r1 * r1;                  // x^2+y^2  or  z^2
    float cp = c0 * c0 + c1 * c1;
    const float rnorm = rp + __shfl_xor(rp, 16, 32);   // |row l15|^2
    const float cnorm = cp + __shfl_xor(cp, 16, 32);   // |col l15|^2

    // Fragment 1: A=(-2x,-2y | -2z, rnorm),  B=(x,y | z, 1)
    v2f a1, b1;
    a1.x = -2.0f * r0;
    a1.y = (half == 0) ? (-2.0f * r1) : rnorm;
    b1.x = c0;
    b1.y = (half == 0) ? c1 : 1.0f;

    // Fragment 2: A=(1,0 | 0,0),  B=(cnorm,0 | 0,0)
    v2f a2, b2;
    a2.x = (half == 0) ? 1.0f : 0.0f;
    a2.y = 0.0f;
    b2.x = (half == 0) ? cnorm : 0.0f;
    b2.y = 0.0f;

    v8f c = {};
    c = __builtin_amdgcn_wmma_f32_16x16x4_f32(false, a1, false, b1,
                                              (short)0, c, false, false);
    c = __builtin_amdgcn_wmma_f32_16x16x4_f32(false, a2, false, b2,
                                              (short)0, c, false, false);

    // Per-plane base pointers; row offsets fold into store IOFFSET (8KB*v).
    const size_t idx0 = ((size_t)b * NPTS + (size_t)(tr * 16 + half * 8)) * NPTS
                      + (size_t)tc * 16 + l15;
    float* __restrict__ o3 = out + idx0;
    float* __restrict__ o5 = out + PLANE + idx0;
    float* __restrict__ o8 = out + 2 * PLANE + idx0;

    #pragma unroll
    for (int v = 0; v < 8; ++v) {
        const float sq = c[v];                       // dist^2 (may be ~-eps)
        const float dist = __builtin_amdgcn_sqrtf(fmaxf(sq, 1e-12f));
        const bool live = sq > 1e-12f;               // dist > 1e-6
        const float w3 = (live && sq <  9.0f) ? dist : 0.0f;
        const float w5 = (live && sq < 25.0f) ? dist : 0.0f;
        const float w8 = (live && sq < 64.0f) ? dist : 0.0f;

        __builtin_nontemporal_store(w3, o3 + (size_t)v * NPTS);
        __builtin_nontemporal_store(w5, o5 + (size_t)v * NPTS);
        __builtin_nontemporal_store(w8, o8 + (size_t)v * NPTS);
    }
}

extern "C" void kernel_launch(void* const* d_in, const int* in_sizes, int n_in,
                              void* d_out, int out_size, void* d_ws, size_t ws_size,
                              hipStream_t stream)
{
    (void)in_sizes; (void)n_in; (void)d_ws; (void)ws_size; (void)out_size;
    const float* pos = (const float*)d_in[0];
    float* out = (float*)d_out;

    const int total_tiles = NBATCH * NTILE * NTILE;   // 131072
    const int blocks = total_tiles / 8;               // 8 waves / 256-thread block
    msgc_dist_kernel<<<blocks, 256, 0, stream>>>(pos, out);
}